// GRN_60705067762110
// MI455X (gfx1250) — compile-verified
//
#include <hip/hip_runtime.h>

typedef __attribute__((ext_vector_type(2))) float v2f;
typedef __attribute__((ext_vector_type(8))) float v8f;

#define NNODES   20000
#define KNBR     32
#define FIN      128
#define FOUT     128
#define TILE_N   32            // nodes per block
#define LDS_STRIDE 132         // 128 + 4 floats pad: (4*M + k) % 64 distinct -> no bank conflicts

__global__ __launch_bounds__(256) void grn_wmma_kernel(
    const float* __restrict__ neighbors,   // [N, K, FIN]
    const float* __restrict__ attn,        // [N, K]
    const float* __restrict__ W,           // [FOUT, FIN]
    const float* __restrict__ bias,        // [FOUT]
    float* __restrict__ out)               // [N, FOUT]
{
    __shared__ __align__(16) float lds_agg[TILE_N * LDS_STRIDE];

    const int tid       = threadIdx.x;
    const int blockNode = blockIdx.x * TILE_N;

    // ------------------------------------------------------------------
    // Phase 1: attention-weighted aggregation (the bandwidth-bound part).
    // 8 threads per node; each thread owns 16 consecutive features.
    // Per (node,k): 8 threads issue coalesced 512B of b128 loads.
    // ------------------------------------------------------------------
    {
        const int nl    = tid >> 3;     // local node 0..31
        const int chunk = tid & 7;      // which 16-float slice of FIN
        const long long nglob = (long long)blockNode + nl;

        const float* nbase = neighbors + nglob * (long long)(KNBR * FIN) + chunk * 16;
        const float* abase = attn + nglob * KNBR;

        float4 c0 = make_float4(0.f, 0.f, 0.f, 0.f);
        float4 c1 = c0, c2 = c0, c3 = c0;

        #pragma unroll 4
        for (int k = 0; k < KNBR; ++k) {
            const float a = abase[k];
            const float4* p = (const float4*)(nbase + (long long)k * FIN);
            float4 x0 = p[0], x1 = p[1], x2 = p[2], x3 = p[3];
            c0.x = fmaf(a, x0.x, c0.x); c0.y = fmaf(a, x0.y, c0.y);
            c0.z = fmaf(a, x0.z, c0.z); c0.w = fmaf(a, x0.w, c0.w);
            c1.x = fmaf(a, x1.x, c1.x); c1.y = fmaf(a, x1.y, c1.y);
            c1.z = fmaf(a, x1.z, c1.z); c1.w = fmaf(a, x1.w, c1.w);
            c2.x = fmaf(a, x2.x, c2.x); c2.y = fmaf(a, x2.y, c2.y);
            c2.z = fmaf(a, x2.z, c2.z); c2.w = fmaf(a, x2.w, c2.w);
            c3.x = fmaf(a, x3.x, c3.x); c3.y = fmaf(a, x3.y, c3.y);
            c3.z = fmaf(a, x3.z, c3.z); c3.w = fmaf(a, x3.w, c3.w);
        }

        float4* dst = (float4*)&lds_agg[nl * LDS_STRIDE + chunk * 16];
        dst[0] = c0; dst[1] = c1; dst[2] = c2; dst[3] = c3;
    }

    __syncthreads();

    // ------------------------------------------------------------------
    // Phase 2: out = agg @ W^T via V_WMMA_F32_16X16X4_F32.
    // Wave w owns out-feature column tile w (16 cols) for BOTH 16-node
    // row tiles -> B fragment loaded once, two WMMAs per K-step.
    // ------------------------------------------------------------------
    const int wave = tid >> 5;       // 0..7  -> column tile
    const int lane = tid & 31;
    const int mn    = lane & 15;     // A row within tile / C-D column N
    const int khalf = (lane >> 4) * 2;  // K sub-offset {0,2} per ISA A/B f32 layout

    // C init = bias (broadcast along M; N = lane&15)
    const float bv = bias[wave * 16 + mn];
    v8f acc0 = { bv, bv, bv, bv, bv, bv, bv, bv };   // row tile 0 (nodes 0..15)
    v8f acc1 = acc0;                                 // row tile 1 (nodes 16..31)

    const float* lA0 = &lds_agg[mn * LDS_STRIDE];
    const float* lA1 = &lds_agg[(16 + mn) * LDS_STRIDE];
    const float* wB  = W + (long long)(wave * 16 + mn) * FIN;

    #pragma unroll 8
    for (int kk = 0; kk < FIN / 4; ++kk) {
        const int kc = kk * 4 + khalf;
        // A fragments (v2f from padded LDS, 8B aligned, conflict-free)
        v2f a0 = *(const v2f*)(lA0 + kc);
        v2f a1 = *(const v2f*)(lA1 + kc);
        // B fragment: B[k][N] = W[colTile*16 + N][kc + k]  (8B global load, L2-hot)
        v2f b  = *(const v2f*)(wB + kc);

        acc0 = __builtin_amdgcn_wmma_f32_16x16x4_f32(
                   false, a0, false, b, (short)0, acc0, false, false);
        acc1 = __builtin_amdgcn_wmma_f32_16x16x4_f32(
                   false, a1, false, b, (short)0, acc1, false, false);
    }

    // ------------------------------------------------------------------
    // Phase 3: ELU + store. C/D layout: VGPR i -> M = i (lanes 0..15)
    //                                   or M = i+8 (lanes 16..31); N = lane&15.
    // ------------------------------------------------------------------
    const int mbase = (lane < 16) ? 0 : 8;
    const int col   = wave * 16 + mn;

    #pragma unroll
    for (int i = 0; i < 8; ++i) {
        const int m0 = mbase + i;
        float x0 = acc0[i];
        float x1 = acc1[i];
        x0 = (x0 > 0.f) ? x0 : (__expf(x0) - 1.0f);
        x1 = (x1 > 0.f) ? x1 : (__expf(x1) - 1.0f);
        out[(long long)(blockNode + m0)      * FOUT + col] = x0;
        out[(long long)(blockNode + 16 + m0) * FOUT + col] = x1;
    }
}

extern "C" void kernel_launch(void* const* d_in, const int* in_sizes, int n_in,
                              void* d_out, int out_size, void* d_ws, size_t ws_size,
                              hipStream_t stream) {
    // setup_inputs() order: nodes, neighbors, attention_scores, W, bias
    // (nodes is unused by the reference output)
    const float* neighbors = (const float*)d_in[1];
    const float* attn      = (const float*)d_in[2];
    const float* W         = (const float*)d_in[3];
    const float* bias      = (const float*)d_in[4];
    float*       out       = (float*)d_out;

    const int nblocks = NNODES / TILE_N;   // 20000 / 32 = 625 exactly
    grn_wmma_kernel<<<nblocks, 256, 0, stream>>>(neighbors, attn, W, bias, out);
}